// LightGCL_44452911514164
// MI455X (gfx1250) — compile-verified
//
#include <hip/hip_runtime.h>
#include <math.h>

// ---- problem constants (match reference) ----
#define NU      100000
#define NI      50000
#define D_DIM   64
#define Q_DIM   5
#define NNZ_E   1000000
#define BB      512
#define INDIM   768
#define TEMP_F  0.2f
#define LAM1    0.2f
#define LAM2    1e-7f
#define LAM3    0.01f

typedef __attribute__((ext_vector_type(2))) float v2f;
typedef __attribute__((ext_vector_type(8))) float v8f;

// ---------------------------------------------------------------------------
// SpMM scatter: Z[rows[e]] += vals[e] * Eprev[cols[e]]   (4 edges / 256-thr block)
// ---------------------------------------------------------------------------
__global__ void spmm_kernel(const int* __restrict__ rows, const int* __restrict__ cols,
                            const float* __restrict__ vals, const float* __restrict__ Eprev,
                            float* __restrict__ Z, int nnz) {
    int e = blockIdx.x * 4 + (threadIdx.x >> 6);
    int d = threadIdx.x & 63;
    if (e < nnz) {
        float v = vals[e];
        int r = rows[e], c = cols[e];
        atomicAdd(&Z[(size_t)r * D_DIM + d], v * Eprev[(size_t)c * D_DIM + d]);
    }
}

// ---------------------------------------------------------------------------
// T[q][d] = sum_n vt[q][n] * E[n][d]   (one 256-thr block per (q,d) output)
// ---------------------------------------------------------------------------
__global__ void colreduce_T(const float* __restrict__ vt, const float* __restrict__ E,
                            float* __restrict__ T, int N) {
    int q = blockIdx.x >> 6;
    int d = blockIdx.x & 63;
    float s = 0.f;
    for (int n = threadIdx.x; n < N; n += 256)
        s += vt[(size_t)q * N + n] * E[(size_t)n * D_DIM + d];
    __shared__ float red[8];
    for (int off = 16; off; off >>= 1) s += __shfl_down(s, off, 32);
    if ((threadIdx.x & 31) == 0) red[threadIdx.x >> 5] = s;
    __syncthreads();
    if (threadIdx.x == 0) {
        float t = 0.f;
        for (int i = 0; i < 8; ++i) t += red[i];
        T[q * D_DIM + d] = t;
    }
}

// ---------------------------------------------------------------------------
// Esum += Z ;  Gsum += mul_s @ T     (fused elementwise over N*64)
// ---------------------------------------------------------------------------
__global__ void accum_EG(float* __restrict__ Esum, const float* __restrict__ Z,
                         float* __restrict__ Gsum, const float* __restrict__ mul_s,
                         const float* __restrict__ T, int total) {
    int i = blockIdx.x * 256 + threadIdx.x;
    if (i < total) {
        int r = i >> 6, d = i & 63;
        Esum[i] += Z[i];
        float g = 0.f;
#pragma unroll
        for (int q = 0; q < Q_DIM; ++q)
            g += mul_s[(size_t)r * Q_DIM + q] * T[q * D_DIM + d];
        Gsum[i] += g;
    }
}

__global__ void build_iids(const int* __restrict__ pos, const int* __restrict__ neg,
                           int* __restrict__ iids) {
    int i = blockIdx.x * 256 + threadIdx.x;
    if (i < 2 * BB) iids[i] = (i < BB) ? pos[i] : neg[i - BB];
}

__global__ void gather_rows(const float* __restrict__ src, const int* __restrict__ ids,
                            float* __restrict__ dst, int total) {
    int i = blockIdx.x * 256 + threadIdx.x;
    if (i < total) dst[i] = src[(size_t)ids[i >> 6] * D_DIM + (i & 63)];
}

// ---------------------------------------------------------------------------
// Fused WMMA logsumexp-row-sum:
//   rowsum[m] += sum_n exp( (Gsel[m] . E[n]) / TEMP )
// One wave per 16-row M tile; K=64 via 16 chained v_wmma_f32_16x16x4_f32.
// A frags live in VGPRs for the whole N sweep (logits never touch memory).
// ---------------------------------------------------------------------------
__global__ __launch_bounds__(32) void lse_wmma_kernel(
    const float* __restrict__ Gsel, const float* __restrict__ E,
    int Nrows, int tiles_per_block, float inv_temp, float* __restrict__ rowsum) {
    const int lane = threadIdx.x & 31;
    const int row  = lane & 15;   // M row (A) / N col (B/C)
    const int half = lane >> 4;   // K-half selector for A/B frags; M-half for C
    const int mtile = blockIdx.x;

    // A fragment: 16x4 f32 layout -> lane holds {A[row][2*half], A[row][2*half+1]}
    const float* Arow = Gsel + (size_t)(mtile * 16 + row) * D_DIM;
    v2f a[16];
#pragma unroll
    for (int kk = 0; kk < 16; ++kk) {
        const int k = kk * 4 + half * 2;
        a[kk][0] = Arow[k];
        a[kk][1] = Arow[k + 1];
    }

    float accr[8];
#pragma unroll
    for (int v = 0; v < 8; ++v) accr[v] = 0.f;

    const int ntiles = Nrows >> 4;
    int t0 = blockIdx.y * tiles_per_block;
    int t1 = t0 + tiles_per_block; if (t1 > ntiles) t1 = ntiles;

    for (int t = t0; t < t1; ++t) {
        // B = E.T tile: B[k][col] = E[t*16+col][k] -> same per-lane addressing as A
        const float* Brow = E + (size_t)(t * 16 + row) * D_DIM;
        if (t + 1 < t1) __builtin_prefetch(Brow + 16 * D_DIM, 0, 1);
        v8f c = {};
#pragma unroll
        for (int kk = 0; kk < 16; ++kk) {
            const int k = kk * 4 + half * 2;
            v2f b;
            b[0] = Brow[k];
            b[1] = Brow[k + 1];
            c = __builtin_amdgcn_wmma_f32_16x16x4_f32(false, a[kk], false, b,
                                                      (short)0, c, false, false);
        }
        // c[v] = logits tile at (M = v + 8*half, N-col = row); fold exp into row acc
#pragma unroll
        for (int v = 0; v < 8; ++v)
            accr[v] += __expf(c[v] * inv_temp);
    }

    // reduce the 16 N-columns held across lanes of each half
#pragma unroll
    for (int off = 1; off < 16; off <<= 1) {
#pragma unroll
        for (int v = 0; v < 8; ++v)
            accr[v] += __shfl_xor(accr[v], off, 32);
    }
    if (row == 0) {
#pragma unroll
        for (int v = 0; v < 8; ++v)
            atomicAdd(&rowsum[mtile * 16 + half * 8 + v], accr[v]);
    }
}

// ---------------------------------------------------------------------------
// clip(dot(Gsel[b], E[ids[b]])/TEMP, -5, 5) accumulated into *acc
// ---------------------------------------------------------------------------
__global__ __launch_bounds__(32) void dot_clip_kernel(
    const float* __restrict__ Gsel, const float* __restrict__ E,
    const int* __restrict__ ids, float inv_temp, float* __restrict__ acc) {
    int b = blockIdx.x, lane = threadIdx.x;
    const float* g = Gsel + (size_t)b * D_DIM;
    const float* e = E + (size_t)ids[b] * D_DIM;
    float s = g[lane] * e[lane] + g[lane + 32] * e[lane + 32];
    for (int off = 16; off; off >>= 1) s += __shfl_xor(s, off, 32);
    if (lane == 0) {
        float v = fminf(fmaxf(s * inv_temp, -5.f), 5.f);
        atomicAdd(acc, v);
    }
}

__global__ __launch_bounds__(32) void bpr_kernel(
    const float* __restrict__ Eu, const float* __restrict__ Ei,
    const int* __restrict__ uids, const int* __restrict__ pos,
    const int* __restrict__ neg, float* __restrict__ acc) {
    int b = blockIdx.x, lane = threadIdx.x;
    const float* u = Eu + (size_t)uids[b] * D_DIM;
    const float* p = Ei + (size_t)pos[b]  * D_DIM;
    const float* n = Ei + (size_t)neg[b]  * D_DIM;
    float ps = u[lane] * p[lane] + u[lane + 32] * p[lane + 32];
    float ns = u[lane] * n[lane] + u[lane + 32] * n[lane + 32];
    for (int off = 16; off; off >>= 1) {
        ps += __shfl_xor(ps, off, 32);
        ns += __shfl_xor(ns, off, 32);
    }
    if (lane == 0) {
        float x = ps - ns;
        float ls = fminf(x, 0.f) - log1pf(__expf(-fabsf(x)));  // log_sigmoid(x)
        atomicAdd(acc, ls);
    }
}

__global__ void sumsq_kernel(const float* __restrict__ x, int n, float* __restrict__ acc) {
    float s = 0.f;
    for (int i = blockIdx.x * blockDim.x + threadIdx.x; i < n; i += gridDim.x * blockDim.x) {
        float v = x[i];
        s += v * v;
    }
    for (int off = 16; off; off >>= 1) s += __shfl_down(s, off, 32);
    if ((threadIdx.x & 31) == 0) atomicAdd(acc, s);
}

// 64 users x 8 unpop x 32 pop pairwise L2 distances, each / n_pop (=32)
__global__ __launch_bounds__(32) void extra_kernel(
    const float* __restrict__ Ei, const int* __restrict__ unpop,
    const int* __restrict__ pop, float* __restrict__ acc) {
    int pair = blockIdx.x;                 // u*256 + a*32 + p
    int u = pair >> 8, a = (pair >> 5) & 7, p = pair & 31;
    int lane = threadIdx.x;
    const float* x = Ei + (size_t)unpop[u * 8 + a] * D_DIM;
    const float* y = Ei + (size_t)pop[u * 32 + p] * D_DIM;
    float d0 = x[lane] - y[lane];
    float d1 = x[lane + 32] - y[lane + 32];
    float s = d0 * d0 + d1 * d1;
    for (int off = 16; off; off >>= 1) s += __shfl_xor(s, off, 32);
    if (lane == 0) atomicAdd(acc, sqrtf(s) * (1.0f / 32.0f));
}

__global__ void finalize_kernel(const float* __restrict__ rsu, const float* __restrict__ rsi,
                                const float* __restrict__ acc, float* __restrict__ out) {
    float nu = 0.f;
    for (int b = 0; b < BB; ++b) nu += logf(rsu[b] + 1e-8f);
    nu /= (float)BB;
    float ni = 0.f;
    for (int b = 0; b < 2 * BB; ++b) ni += logf(rsi[b] + 1e-8f);
    ni /= (float)(2 * BB);
    float neg_score = nu + ni;
    float pos_score = acc[0] / (float)BB + acc[1] / (float)(2 * BB);
    float loss_s = neg_score - pos_score;
    float loss_r = -(acc[2] / (float)BB);
    float loss_reg = LAM2 * acc[3];
    float loss = loss_r + LAM1 * loss_s + loss_reg + LAM3 * acc[4];
    out[0] = loss;
    out[1] = loss_r;
    out[2] = LAM1 * loss_s;
}

// ---------------------------------------------------------------------------
extern "C" void kernel_launch(void* const* d_in, const int* in_sizes, int n_in,
                              void* d_out, int out_size, void* d_ws, size_t ws_size,
                              hipStream_t stream) {
    (void)in_sizes; (void)n_in; (void)out_size; (void)ws_size;

    const int*   uids      = (const int*)d_in[0];
    const int*   pos       = (const int*)d_in[1];
    const int*   neg       = (const int*)d_in[2];
    const int*   adj_rows  = (const int*)d_in[3];
    const int*   adj_cols  = (const int*)d_in[4];
    const float* adj_vals  = (const float*)d_in[5];
    const float* E_u_0     = (const float*)d_in[6];
    const float* E_i_0     = (const float*)d_in[7];
    const float* u_mul_s   = (const float*)d_in[8];
    const float* vt        = (const float*)d_in[9];
    const float* v_mul_s   = (const float*)d_in[10];
    const float* ut        = (const float*)d_in[11];
    const float* W_api     = (const float*)d_in[12];
    const float* b_api     = (const float*)d_in[13];
    const float* W_mashup  = (const float*)d_in[14];
    const float* b_mashup  = (const float*)d_in[15];
    const float* pos_api   = (const float*)d_in[16];
    const float* neg_api   = (const float*)d_in[17];
    const float* mashup    = (const float*)d_in[18];
    const int*   unpop_idx = (const int*)d_in[19];
    const int*   pop_idx   = (const int*)d_in[20];

    float* out = (float*)d_out;
    float* ws  = (float*)d_ws;

    // output layout: [loss, loss_r, lam1*loss_s, mashup, pos_api, neg_api, E_u, E_i]
    const size_t OFF_MASH = 3;
    const size_t OFF_PAPI = OFF_MASH + (size_t)BB * INDIM;
    const size_t OFF_NAPI = OFF_PAPI + (size_t)BB * INDIM;
    const size_t OFF_EU   = OFF_NAPI + (size_t)BB * INDIM;
    const size_t OFF_EI   = OFF_EU + (size_t)NU * D_DIM;
    float* Eu = out + OFF_EU;
    float* Ei = out + OFF_EI;

    // workspace layout (float offsets)
    const size_t ZU0 = 0;
    const size_t ZU1 = ZU0 + (size_t)NU * D_DIM;
    const size_t ZI0 = ZU1 + (size_t)NU * D_DIM;
    const size_t ZI1 = ZI0 + (size_t)NI * D_DIM;
    const size_t GU  = ZI1 + (size_t)NI * D_DIM;
    const size_t GI  = GU + (size_t)NU * D_DIM;
    const size_t TU  = GI + (size_t)NI * D_DIM;
    const size_t TI  = TU + Q_DIM * D_DIM;
    const size_t GUS = TI + Q_DIM * D_DIM;
    const size_t GIS = GUS + (size_t)BB * D_DIM;
    const size_t RSU = GIS + (size_t)2 * BB * D_DIM;
    const size_t RSI = RSU + BB;
    const size_t ACC = RSI + 2 * BB;
    const size_t IID = ACC + 16;

    // pass-through outputs
    hipMemcpyAsync(out + OFF_MASH, mashup,  (size_t)BB * INDIM * sizeof(float), hipMemcpyDeviceToDevice, stream);
    hipMemcpyAsync(out + OFF_PAPI, pos_api, (size_t)BB * INDIM * sizeof(float), hipMemcpyDeviceToDevice, stream);
    hipMemcpyAsync(out + OFF_NAPI, neg_api, (size_t)BB * INDIM * sizeof(float), hipMemcpyDeviceToDevice, stream);

    // init sums with layer-0 embeddings
    hipMemcpyAsync(Eu,      E_u_0, (size_t)NU * D_DIM * sizeof(float), hipMemcpyDeviceToDevice, stream);
    hipMemcpyAsync(Ei,      E_i_0, (size_t)NI * D_DIM * sizeof(float), hipMemcpyDeviceToDevice, stream);
    hipMemcpyAsync(ws + GU, E_u_0, (size_t)NU * D_DIM * sizeof(float), hipMemcpyDeviceToDevice, stream);
    hipMemcpyAsync(ws + GI, E_i_0, (size_t)NI * D_DIM * sizeof(float), hipMemcpyDeviceToDevice, stream);

    // zero atomic targets
    hipMemsetAsync(ws + ZU0, 0, (size_t)(2 * NU * D_DIM + 2 * NI * D_DIM) * sizeof(float), stream);
    hipMemsetAsync(ws + RSU, 0, (size_t)(BB + 2 * BB + 16) * sizeof(float), stream);

    // L = 2 propagation layers
    for (int layer = 0; layer < 2; ++layer) {
        const float* Eu_prev = (layer == 0) ? E_u_0 : ws + ZU0;
        const float* Ei_prev = (layer == 0) ? E_i_0 : ws + ZI0;
        float* Zu = (layer == 0) ? ws + ZU0 : ws + ZU1;
        float* Zi = (layer == 0) ? ws + ZI0 : ws + ZI1;

        spmm_kernel<<<NNZ_E / 4, 256, 0, stream>>>(adj_rows, adj_cols, adj_vals, Ei_prev, Zu, NNZ_E);
        spmm_kernel<<<NNZ_E / 4, 256, 0, stream>>>(adj_cols, adj_rows, adj_vals, Eu_prev, Zi, NNZ_E);
        colreduce_T<<<Q_DIM * D_DIM, 256, 0, stream>>>(vt, Ei_prev, ws + TU, NI);
        colreduce_T<<<Q_DIM * D_DIM, 256, 0, stream>>>(ut, Eu_prev, ws + TI, NU);
        accum_EG<<<(NU * D_DIM + 255) / 256, 256, 0, stream>>>(Eu, Zu, ws + GU, u_mul_s, ws + TU, NU * D_DIM);
        accum_EG<<<(NI * D_DIM + 255) / 256, 256, 0, stream>>>(Ei, Zi, ws + GI, v_mul_s, ws + TI, NI * D_DIM);
    }

    // gather batch rows of G sums
    int* iids = (int*)(ws + IID);
    build_iids<<<4, 256, 0, stream>>>(pos, neg, iids);
    gather_rows<<<(BB * D_DIM + 255) / 256, 256, 0, stream>>>(ws + GU, uids, ws + GUS, BB * D_DIM);
    gather_rows<<<(2 * BB * D_DIM + 255) / 256, 256, 0, stream>>>(ws + GI, iids, ws + GIS, 2 * BB * D_DIM);

    // fused WMMA logits + exp + row-sum (logsumexp denominators)
    const float inv_temp = 1.0f / TEMP_F;
    lse_wmma_kernel<<<dim3(BB / 16, 125), 32, 0, stream>>>(ws + GUS, Eu, NU, (NU / 16) / 125, inv_temp, ws + RSU);
    lse_wmma_kernel<<<dim3(2 * BB / 16, 125), 32, 0, stream>>>(ws + GIS, Ei, NI, (NI / 16) / 125, inv_temp, ws + RSI);

    // positive terms, BPR, regularization, popularity distances
    dot_clip_kernel<<<BB, 32, 0, stream>>>(ws + GUS, Eu, uids, inv_temp, ws + ACC + 0);
    dot_clip_kernel<<<2 * BB, 32, 0, stream>>>(ws + GIS, Ei, iids, inv_temp, ws + ACC + 1);
    bpr_kernel<<<BB, 32, 0, stream>>>(Eu, Ei, uids, pos, neg, ws + ACC + 2);
    sumsq_kernel<<<512, 256, 0, stream>>>(E_u_0, NU * D_DIM, ws + ACC + 3);
    sumsq_kernel<<<256, 256, 0, stream>>>(E_i_0, NI * D_DIM, ws + ACC + 3);
    sumsq_kernel<<<64, 256, 0, stream>>>(W_api, INDIM * D_DIM, ws + ACC + 3);
    sumsq_kernel<<<1, 256, 0, stream>>>(b_api, D_DIM, ws + ACC + 3);
    sumsq_kernel<<<64, 256, 0, stream>>>(W_mashup, INDIM * D_DIM, ws + ACC + 3);
    sumsq_kernel<<<1, 256, 0, stream>>>(b_mashup, D_DIM, ws + ACC + 3);
    extra_kernel<<<64 * 8 * 32, 32, 0, stream>>>(Ei, unpop_idx, pop_idx, ws + ACC + 4);

    finalize_kernel<<<1, 1, 0, stream>>>(ws + RSU, ws + RSI, ws + ACC, out);
}